// NnHalfKA_36498632081978
// MI455X (gfx1250) — compile-verified
//
#include <hip/hip_runtime.h>
#include <hip/hip_bf16.h>
#include <math.h>

typedef __attribute__((ext_vector_type(2))) float v2f;
typedef __attribute__((ext_vector_type(8))) float v8f;

#define FTW      512      // feature-transform width
#define F_VIRT   768
#define TILE_M   16       // batch rows per block (one wave32 each)
#define LDS_PITCH (FTW + 4)   // pad 4 floats -> stride 516 (516 % 64 == 4), conflict-free

__device__ __forceinline__ int lower_bound_i32(const int* __restrict__ a, int n, int key) {
    int lo = 0, hi = n;
    while (lo < hi) {
        int mid = (lo + hi) >> 1;
        if (a[mid] < key) lo = mid + 1; else hi = mid;
    }
    return lo;
}

__device__ __forceinline__ void fma4(float4& d, float v, const float4 a, const float4 b) {
    d.x = fmaf(v, a.x + b.x, d.x);
    d.y = fmaf(v, a.y + b.y, d.y);
    d.z = fmaf(v, a.z + b.z, d.z);
    d.w = fmaf(v, a.w + b.w, d.w);
}

__device__ __forceinline__ float clamp01(float x) { return fminf(fmaxf(x, 0.0f), 1.0f); }

// One pass of the 16-row GEMV over a 512-wide segment of W_out using fp32 WMMA.
// A: 16x4 tile of hidden rows from LDS. B: 4x16 with w[K+k] broadcast across N,
// so D[m][n] == partial dot of row m (identical for every n; we read n==0).
// A layout (32-bit 16x4): lanes 0-15 hold K=0,1 ; lanes 16-31 hold K=2,3.
// B layout (32-bit 4x16, C-style striping): VGPR0 = rows K=0|K=2, VGPR1 = rows K=1|K=3.
__device__ __forceinline__ void gemv_pass_wmma(const float (&h)[TILE_M][LDS_PITCH],
                                               const float* __restrict__ wseg,
                                               int lane, v8f& c0, v8f& c1)
{
    const int m    = lane & 15;
    const int koff = (lane < 16) ? 0 : 2;
#pragma unroll 4
    for (int K = 0; K < FTW; K += 8) {
        v2f a0 = *(const v2f*)&h[m][K + koff];
        v2f b0 = *(const v2f*)(wseg + K + koff);
        c0 = __builtin_amdgcn_wmma_f32_16x16x4_f32(false, a0, false, b0,
                                                   (short)0, c0, false, false);
        v2f a1 = *(const v2f*)&h[m][K + 4 + koff];
        v2f b1 = *(const v2f*)(wseg + K + 4 + koff);
        c1 = __builtin_amdgcn_wmma_f32_16x16x4_f32(false, a1, false, b1,
                                                   (short)0, c1, false, false);
    }
}

__global__ __launch_bounds__(TILE_M * 32)
void nnue_halfka_kernel(const float* __restrict__ values,
                        const float* __restrict__ W_ft,
                        const float* __restrict__ ft_b,
                        const float* __restrict__ W_fft,
                        const float* __restrict__ fft_b,
                        const float* __restrict__ W_out,
                        const float* __restrict__ out_b,
                        const int*   __restrict__ batch_ids,
                        const int*   __restrict__ stm_feat,
                        const int*   __restrict__ nstm_feat,
                        float*       __restrict__ out,
                        int nnz, int bsize)
{
    __shared__ __align__(16) float lds_h[TILE_M][LDS_PITCH];

    const int tid  = threadIdx.x;
    const int lane = tid & 31;
    const int row  = tid >> 5;                                    // wave id == row in tile
    const int b    = __builtin_amdgcn_readfirstlane((int)(blockIdx.x * TILE_M + row));

    // Sorted-COO segment for this batch element (wave-uniform -> scalar path).
    const int s0 = lower_bound_i32(batch_ids, nnz, b);
    const int s1 = lower_bound_i32(batch_ids, nnz, b + 1);

    float4 acc_s[4], acc_n[4];
#pragma unroll
    for (int j = 0; j < 4; ++j) {
        acc_s[j] = make_float4(0.f, 0.f, 0.f, 0.f);
        acc_n[j] = make_float4(0.f, 0.f, 0.f, 0.f);
    }

    // Process the segment in chunks of 32 nonzeros: one coalesced load per
    // metadata array per chunk, then broadcast each nonzero's (v, sf, nf) via
    // v_readlane (SGPR result -> scalar row bases, GVS addressing for the
    // gathers; no memory-latency-carried dependency in the hot loop).
    for (int base = s0; base < s1; base += 32) {
        const int i     = min(base + lane, s1 - 1);    // clamped, always in-range
        const uint vl   = __float_as_uint(values[i]);
        const int  sfl  = stm_feat[i];
        const int  nfl  = nstm_feat[i];
        const int  cnt  = min(32, s1 - base);

        for (int t = 0; t < cnt; ++t) {
            const float v  = __uint_as_float(__builtin_amdgcn_readlane(vl, t));
            const int   sf = __builtin_amdgcn_readlane(sfl, t);
            const int   nf = __builtin_amdgcn_readlane(nfl, t);

            const float4* rs  = (const float4*)(W_ft  + (size_t)sf * FTW);
            const float4* rsv = (const float4*)(W_fft + (size_t)(sf % F_VIRT) * FTW);
            const float4* rn  = (const float4*)(W_ft  + (size_t)nf * FTW);
            const float4* rnv = (const float4*)(W_fft + (size_t)(nf % F_VIRT) * FTW);
#pragma unroll
            for (int j = 0; j < 4; ++j) {
                const int e = j * 32 + lane;           // float4 index (col = 4*e)
                fma4(acc_s[j], v, rs[e], rsv[e]);
                fma4(acc_n[j], v, rn[e], rnv[e]);
            }
        }
    }

    // Bias per owned column chunk (lane owns cols j*128 + lane*4 .. +3).
    float4 bia[4];
#pragma unroll
    for (int j = 0; j < 4; ++j) {
        const int c0col = j * 128 + lane * 4;
        const float4 fb = *(const float4*)(ft_b  + c0col);
        const float4 vb = *(const float4*)(fft_b + c0col);
        bia[j] = make_float4(fb.x + vb.x, fb.y + vb.y, fb.z + vb.z, fb.w + vb.w);
    }

    // -------- pass 1: stm half --------
#pragma unroll
    for (int j = 0; j < 4; ++j) {
        const int c0col = j * 128 + lane * 4;
        float4 h;
        h.x = clamp01(acc_s[j].x + bia[j].x);
        h.y = clamp01(acc_s[j].y + bia[j].y);
        h.z = clamp01(acc_s[j].z + bia[j].z);
        h.w = clamp01(acc_s[j].w + bia[j].w);
        *(float4*)&lds_h[row][c0col] = h;
    }
    __syncthreads();

    v8f c0 = {}, c1 = {};
    if (tid < 32) {
        gemv_pass_wmma(lds_h, W_out, lane, c0, c1);
    }
    __syncthreads();

    // -------- pass 2: nstm half --------
#pragma unroll
    for (int j = 0; j < 4; ++j) {
        const int c0col = j * 128 + lane * 4;
        float4 h;
        h.x = clamp01(acc_n[j].x + bia[j].x);
        h.y = clamp01(acc_n[j].y + bia[j].y);
        h.z = clamp01(acc_n[j].z + bia[j].z);
        h.w = clamp01(acc_n[j].w + bia[j].w);
        *(float4*)&lds_h[row][c0col] = h;
    }
    __syncthreads();

    if (tid < 32) {
        gemv_pass_wmma(lds_h, W_out + FTW, lane, c0, c1);
        const v8f c = c0 + c1;
        const float ob = out_b[0];
        // D[m][0]: lane 0 holds M=0..7 in vgprs 0..7, lane 16 holds M=8..15.
        if (lane == 0 || lane == 16) {
            const int mb = (int)(blockIdx.x * TILE_M) + ((lane == 0) ? 0 : 8);
#pragma unroll
            for (int r = 0; r < 8; ++r) {
                if (mb + r < bsize) {
                    const float z = c[r] + ob;
                    out[mb + r] = 1.0f / (1.0f + __expf(-z));
                }
            }
        }
    }
}

extern "C" void kernel_launch(void* const* d_in, const int* in_sizes, int n_in,
                              void* d_out, int out_size, void* d_ws, size_t ws_size,
                              hipStream_t stream) {
    const float* values    = (const float*)d_in[0];
    const float* W_ft      = (const float*)d_in[1];
    const float* ft_b      = (const float*)d_in[2];
    const float* W_fft     = (const float*)d_in[3];
    const float* fft_b     = (const float*)d_in[4];
    const float* W_out     = (const float*)d_in[5];
    const float* out_b     = (const float*)d_in[6];
    const int*   batch_ids = (const int*)d_in[7];
    const int*   stm_feat  = (const int*)d_in[8];
    const int*   nstm_feat = (const int*)d_in[9];
    float*       out       = (float*)d_out;

    const int nnz   = in_sizes[0];
    const int bsize = out_size;            // B == number of outputs

    dim3 grid((bsize + TILE_M - 1) / TILE_M);
    dim3 block(TILE_M * 32);
    hipLaunchKernelGGL(nnue_halfka_kernel, grid, block, 0, stream,
                       values, W_ft, ft_b, W_fft, fft_b, W_out, out_b,
                       batch_ids, stm_feat, nstm_feat, out, nnz, bsize);
}